// DispersionInteraction_20650202759279
// MI455X (gfx1250) — compile-verified
//
#include <hip/hip_runtime.h>
#include <hip/hip_bf16.h>
#include <stdint.h>

// ---------------------------------------------------------------------------
// vdW-QDO dispersion energy, per-edge -> segment-sum over receivers.
//   Kernel 1 (nodes): precompute per-node {alpha_n, c6_n} float2 table in ws
//                     (800KB, L2-resident) and zero the output.
//   Kernel 2 (edges): double-buffered GLOBAL_LOAD_ASYNC_TO_LDS_B128/B64
//                     staging of (sender, receiver, length) tiles; per tile:
//                     gather float2 params (L2), ~50 VALU ops incl. native
//                     v_log/v_exp, then per-lane global_atomic_add_f32.
// HBM traffic ~77MB -> ~3.3us floor at 23.3 TB/s; gathers + atomics hit L2.
// Confirmed lowerings (round 4): global_atomic_add_f32 scope:SCOPE_DEV,
// async-to-LDS ops + s_wait_asynccnt, ds_load of staged tiles.
// ---------------------------------------------------------------------------

#define BLOCK 256

#if defined(__has_builtin)
#if __has_builtin(__builtin_amdgcn_global_load_async_to_lds_b128)
#define HAS_ASYNC_B128 1
#endif
#if __has_builtin(__builtin_amdgcn_global_load_async_to_lds_b64)
#define HAS_ASYNC_B64 1
#endif
#endif
#ifndef HAS_ASYNC_B128
#define HAS_ASYNC_B128 0
#endif
#ifndef HAS_ASYNC_B64
#define HAS_ASYNC_B64 0
#endif

#if HAS_ASYNC_B128
#define HAS_ASYNC_LDS 1
#define EPT 4                   // 16B per lane per async op
#elif HAS_ASYNC_B64
#define HAS_ASYNC_LDS 1
#define EPT 2                   // 8B per lane per async op
#else
#define HAS_ASYNC_LDS 0
#define EPT 1
#endif
#define TILE (BLOCK * EPT)

__constant__ float c_alphas[102] = {
    4.5f, 1.38f, 164.2f, 38.0f, 21.0f, 12.0f, 7.4f, 5.4f, 3.8f, 2.67f,
    162.7f, 71.0f, 60.0f, 37.0f, 25.0f, 19.6f, 15.0f, 11.1f, 292.9f, 160.0f,
    120.0f, 98.0f, 84.0f, 78.0f, 63.0f, 56.0f, 50.0f, 48.0f, 42.0f, 40.0f,
    60.0f, 41.0f, 29.0f, 25.0f, 20.0f, 16.8f, 319.2f, 199.0f, 126.74f, 119.97f,
    101.6f, 88.42f, 80.08f, 65.89f, 56.1f, 23.68f, 50.6f, 39.7f, 70.22f, 55.95f,
    43.67f, 37.65f, 35.0f, 27.3f, 399.9f, 275.0f, 213.7f, 204.7f, 215.8f, 208.4f,
    200.2f, 192.1f, 184.2f, 158.3f, 169.5f, 164.64f, 156.3f, 150.2f, 144.3f, 138.9f,
    137.2f, 99.52f, 82.53f, 71.04f, 63.04f, 55.06f, 42.51f, 39.68f, 36.5f, 33.9f,
    69.92f, 61.8f, 49.02f, 45.01f, 38.93f, 33.54f, 317.8f, 246.2f, 203.3f, 217.0f,
    154.4f, 127.8f, 150.5f, 132.2f, 131.2f, 143.6f, 125.3f, 121.5f, 117.5f, 113.4f,
    109.4f, 105.4f};

__constant__ float c_c6[102] = {
    6.5f, 1.46f, 1387.0f, 214.0f, 99.5f, 46.6f, 24.2f, 15.6f, 9.52f, 6.38f,
    1556.0f, 627.0f, 528.0f, 305.0f, 185.0f, 134.0f, 94.6f, 64.3f, 3897.0f, 2221.0f,
    1383.0f, 1044.0f, 832.0f, 602.0f, 552.0f, 482.0f, 408.0f, 373.0f, 253.0f, 284.0f,
    498.0f, 354.0f, 246.0f, 210.0f, 162.0f, 129.6f, 4691.0f, 3170.0f, 1968.58f, 1677.91f,
    1263.61f, 1028.73f, 1390.87f, 609.75f, 469.0f, 157.5f, 339.0f, 452.0f, 707.05f, 587.42f,
    459.32f, 396.0f, 385.0f, 285.9f, 6846.0f, 5727.0f, 3884.5f, 3708.33f, 3911.84f, 3908.75f,
    3847.68f, 3708.69f, 3511.71f, 2781.53f, 3124.41f, 2984.29f, 2839.95f, 2724.12f, 2576.78f, 2387.53f,
    2371.8f, 1274.8f, 1019.92f, 847.93f, 710.2f, 596.67f, 359.1f, 347.1f, 298.0f, 392.0f,
    717.44f, 697.0f, 571.0f, 530.92f, 457.53f, 390.63f, 4224.44f, 4851.32f, 3604.41f, 4047.54f,
    2876.77f, 2375.89f, 3102.12f, 2820.47f, 2794.0f, 3150.95f, 2756.0f, 2702.57f, 2626.59f, 2548.62f,
    2468.69f, 2386.8f};

// 1/BOHR, FINE_STRUCTURE^(-4/21), 0.5*HARTREE, 1/7
#define INV_BOHR      1.8897261254578281f
#define FS_POW_M4_21  2.5527964f
#define HALF_HARTREE  13.605693122994f
#define ONE_SEVENTH   0.14285714285714285f

// --------------------------- per-edge math core ----------------------------
__device__ __forceinline__ float edge_energy(float ai, float c6i,
                                             float aj, float c6j,
                                             float len) {
  float aij  = 0.5f * (ai + aj);
  float den  = fmaf(ai * ai, c6j, aj * aj * c6i);
  float c6ij = __fdividef(2.0f * c6i * c6j * ai * aj, den);
  // alpha_ij^(1/7) via native v_log_f32 / v_exp_f32 (aij is O(1..400), safe)
  float a7 = __builtin_amdgcn_exp2f(__log2f(aij) * ONE_SEVENTH);
  float vr = FS_POW_M4_21 * a7;
  float s  = fmaf(fmaf(fmaf(-0.00078893f, vr, 0.04125273f), vr,
                       0.24428889f), vr, -0.00433008f);
  float s2  = s * s;
  float c8  = 10.0f * s2 * c6ij;            // (5/gamma)*C6 = 10*s^2*C6
  float c10 = 122.5f * s2 * s2 * c6ij;      // 245/(8*gamma^2)*C6 = 122.5*s^4*C6
  float p   = 5.08f * a7;                   // 2*2.54*alpha^(1/7)
  float R  = len * INV_BOHR;
  float R2 = R * R,  p2 = p * p;
  float R6 = R2 * R2 * R2, p6 = p2 * p2 * p2;
  float R8 = R6 * R2, p8 = p6 * p2;
  float R10 = R8 * R2, p10 = p8 * p2;
  float V3 = -(__fdividef(c6ij, R6 + p6) +
               __fdividef(c8,   R8 + p8) +
               __fdividef(c10,  R10 + p10));
  // smooth switching on [8, 10]
  float c  = (len - 8.0f) * 0.5f;
  float x1 = 1.0f - c;
  float s1 = (x1 > 0.0f) ? __expf(__fdividef(-1.0f, fmaxf(x1, 1e-12f))) : 0.0f;
  float sc = (c  > 0.0f) ? __expf(__fdividef(-1.0f, fmaxf(c,  1e-12f))) : 0.0f;
  float w  = __fdividef(s1, s1 + sc + 1e-12f);
  w = (len > 0.0f) ? w : 0.0f;
  return HALF_HARTREE * V3 * w;             // C_FACTOR * HARTREE
}

// --------- tiny probe (never launched): shows f32 atomic lowering ----------
__global__ void atomic_probe_kernel(float* out, const float* v) {
  __hip_atomic_fetch_add(out, v[threadIdx.x],
                         __ATOMIC_RELAXED, __HIP_MEMORY_SCOPE_AGENT);
}

// -------------------- async-to-LDS staged edge kernel ----------------------
#if HAS_ASYNC_LDS
#if HAS_ASYNC_B128
typedef int vNi __attribute__((vector_size(4 * sizeof(int))));
#else
typedef int vNi __attribute__((vector_size(2 * sizeof(int))));
#endif
typedef __attribute__((address_space(1))) vNi g_vNi;
typedef __attribute__((address_space(3))) vNi l_vNi;

__device__ __forceinline__ void async_copy(const void* g, const void* l) {
  // LDS flat address: low 32 bits are the DS byte address (ISA 10.2).
#if HAS_ASYNC_B128
  __builtin_amdgcn_global_load_async_to_lds_b128(
      (g_vNi*)(unsigned long long)g,
      (l_vNi*)(unsigned int)(unsigned long long)l, 0, 0);
#else
  __builtin_amdgcn_global_load_async_to_lds_b64(
      (g_vNi*)(unsigned long long)g,
      (l_vNi*)(unsigned int)(unsigned long long)l, 0, 0);
#endif
}

__global__ __launch_bounds__(BLOCK)
void edge_kernel_async(const int*    __restrict__ senders,
                       const int*    __restrict__ receivers,
                       const float*  __restrict__ lengths,
                       const float2* __restrict__ params,
                       float*        __restrict__ out,
                       long long n_full, int n_tiles) {
  __shared__ __align__(16) int   ls[2][TILE];
  __shared__ __align__(16) int   lr[2][TILE];
  __shared__ __align__(16) float ll[2][TILE];

  const int tid = threadIdx.x;
  const int e0  = tid * EPT;

  int tile = blockIdx.x;
  const int stride = gridDim.x;
  int buf = 0;

  // prologue: stage first tile (clamped addresses keep EXEC uniform; lanes
  // past n_full load valid-but-unused data and skip processing)
  {
    long long b  = (long long)tile * TILE + e0;
    long long bc = (b > n_full - EPT) ? (n_full - EPT) : b;
    async_copy(senders + bc,   &ls[buf][e0]);
    async_copy(receivers + bc, &lr[buf][e0]);
    async_copy(lengths + bc,   &ll[buf][e0]);
  }

  for (;;) {
    const int next = tile + stride;
    const bool more = next < n_tiles;           // uniform (scalar) branch
    if (more) {
      long long b  = (long long)next * TILE + e0;
      long long bc = (b > n_full - EPT) ? (n_full - EPT) : b;
      async_copy(senders + bc,   &ls[buf ^ 1][e0]);
      async_copy(receivers + bc, &lr[buf ^ 1][e0]);
      async_copy(lengths + bc,   &ll[buf ^ 1][e0]);
      asm volatile("s_wait_asynccnt 0x3" ::: "memory");   // current tile done
    } else {
      asm volatile("s_wait_asynccnt 0x0" ::: "memory");   // everything done
    }

    long long b = (long long)tile * TILE + e0;
    if (b < n_full) {                           // all EPT edges valid (aligned)
      int   sv[EPT], rv[EPT];
      float lv[EPT];
#pragma unroll
      for (int k = 0; k < EPT; ++k) {
        sv[k] = ls[buf][e0 + k];
        rv[k] = lr[buf][e0 + k];
        lv[k] = ll[buf][e0 + k];
      }
#pragma unroll
      for (int k = 0; k < EPT; ++k) {
        __builtin_prefetch(&params[sv[k]], 0, 0);
        __builtin_prefetch(&params[rv[k]], 0, 0);
      }
#pragma unroll
      for (int k = 0; k < EPT; ++k) {
        float2 ps = params[sv[k]], pr = params[rv[k]];
        float en = edge_energy(pr.x, pr.y, ps.x, ps.y, lv[k]);
        __hip_atomic_fetch_add(&out[rv[k]], en,
                               __ATOMIC_RELAXED, __HIP_MEMORY_SCOPE_AGENT);
      }
    }
    if (!more) break;
    tile = next;
    buf ^= 1;
  }
}
#endif  // HAS_ASYNC_LDS

// ---------------- kernel 1: node params + zero output ----------------------
__global__ __launch_bounds__(256)
void node_params_kernel(const float* __restrict__ hirsh,
                        const int*   __restrict__ znum,
                        float2*      __restrict__ params,
                        float*       __restrict__ out,
                        int n_nodes) {
  int i = blockIdx.x * blockDim.x + threadIdx.x;
  if (i >= n_nodes) return;
  int z = znum[i] - 1;
  z = max(0, min(z, 101));
  float h = hirsh[i];
  float2 pr;
  pr.x = c_alphas[z] * h;
  pr.y = c_c6[z] * h * h;
  params[i] = pr;
  out[i] = 0.0f;                 // d_out is poisoned by harness -> must zero
}

// ---------------- direct edge kernel (fallback + tail) ---------------------
__global__ __launch_bounds__(256)
void edge_kernel(const int*    __restrict__ senders,
                 const int*    __restrict__ receivers,
                 const float*  __restrict__ lengths,
                 const float2* __restrict__ params,
                 float*        __restrict__ out,
                 int start, int n_edges) {
  int i = start + blockIdx.x * blockDim.x + threadIdx.x;
  if (i >= n_edges) return;
  int js = senders[i];
  int jr = receivers[i];
  float len = lengths[i];
  __builtin_prefetch(&params[js], 0, 0);
  __builtin_prefetch(&params[jr], 0, 0);
  float2 pj = params[js];
  float2 pi = params[jr];
  float e = edge_energy(pi.x, pi.y, pj.x, pj.y, len);
  __hip_atomic_fetch_add(&out[jr], e, __ATOMIC_RELAXED, __HIP_MEMORY_SCOPE_AGENT);
}

// ---------------- fallback: fused edge kernel (no workspace) ---------------
__global__ __launch_bounds__(256)
void zero_out_kernel(float* __restrict__ out, int n) {
  int i = blockIdx.x * blockDim.x + threadIdx.x;
  if (i < n) out[i] = 0.0f;
}

__global__ __launch_bounds__(256)
void edge_kernel_fused(const int*   __restrict__ senders,
                       const int*   __restrict__ receivers,
                       const float* __restrict__ lengths,
                       const float* __restrict__ hirsh,
                       const int*   __restrict__ znum,
                       float*       __restrict__ out,
                       int n_edges) {
  int i = blockIdx.x * blockDim.x + threadIdx.x;
  if (i >= n_edges) return;
  int js = senders[i];
  int jr = receivers[i];
  float len = lengths[i];
  int zi = max(0, min(znum[jr] - 1, 101));
  int zj = max(0, min(znum[js] - 1, 101));
  float hi = hirsh[jr], hj = hirsh[js];
  float ai = c_alphas[zi] * hi, c6i = c_c6[zi] * hi * hi;
  float aj = c_alphas[zj] * hj, c6j = c_c6[zj] * hj * hj;
  float e = edge_energy(ai, c6i, aj, c6j, len);
  __hip_atomic_fetch_add(&out[jr], e, __ATOMIC_RELAXED, __HIP_MEMORY_SCOPE_AGENT);
}

// ------------------------------- launcher ----------------------------------
extern "C" void kernel_launch(void* const* d_in, const int* in_sizes, int n_in,
                              void* d_out, int out_size, void* d_ws, size_t ws_size,
                              hipStream_t stream) {
  const float* hirsh     = (const float*)d_in[0];
  const int*   znum      = (const int*)  d_in[1];
  const int*   senders   = (const int*)  d_in[2];
  const int*   receivers = (const int*)  d_in[3];
  const float* lengths   = (const float*)d_in[4];
  float* out = (float*)d_out;

  const int n_nodes = in_sizes[0];
  const int n_edges = in_sizes[2];
  const int bs = 256;
  const int nblk = (n_nodes + bs - 1) / bs;

  if (ws_size >= (size_t)n_nodes * sizeof(float2)) {
    float2* params = (float2*)d_ws;
    node_params_kernel<<<nblk, bs, 0, stream>>>(hirsh, znum, params, out, n_nodes);
#if HAS_ASYNC_LDS
    const long long n_full = (long long)(n_edges / EPT) * EPT;
    const int n_tiles = (int)((n_full + TILE - 1) / TILE);
    if (n_tiles > 0) {
      int grid = n_tiles < 4096 ? n_tiles : 4096;
      edge_kernel_async<<<grid, BLOCK, 0, stream>>>(senders, receivers, lengths,
                                                    params, out, n_full, n_tiles);
    }
    if ((long long)n_edges > n_full) {
      edge_kernel<<<1, bs, 0, stream>>>(senders, receivers, lengths, params,
                                        out, (int)n_full, n_edges);
    }
#else
    const int eblk = (n_edges + bs - 1) / bs;
    edge_kernel<<<eblk, bs, 0, stream>>>(senders, receivers, lengths, params,
                                         out, 0, n_edges);
#endif
  } else {
    const int eblk = (n_edges + bs - 1) / bs;
    zero_out_kernel<<<nblk, bs, 0, stream>>>(out, n_nodes);
    edge_kernel_fused<<<eblk, bs, 0, stream>>>(senders, receivers, lengths,
                                               hirsh, znum, out, n_edges);
  }
}